// TopKPooling_43138651521386
// MI455X (gfx1250) — compile-verified
//
#include <hip/hip_runtime.h>
#include <stdint.h>

// ---------------------------------------------------------------------------
// TopK pooling, MI455X (gfx1250). Memory-bound: ~512 MiB min traffic -> ~22us
// at 23.3 TB/s. No matmul -> no WMMA; the CDNA5 path is the Tensor Data Mover
// (tensor_load_to_lds + s_wait_tensorcnt) double-buffering strided tiles into
// LDS, plus non-temporal B128 stores for the 256 MiB mask zero-fill so the
// streaming writes don't pollute the 192 MB L2.
// ---------------------------------------------------------------------------

static constexpr int BB = 32;      // batch
static constexpr int PP = 4096;    // patches
static constexpr int CC = 512;     // concepts
static constexpr int KK = 16;      // top-k

static constexpr int PS     = 16;             // P splits across blocks
static constexpr int PROWS  = PP / PS;        // 256 rows per block
static constexpr int CT     = 2;              // column tiles
static constexpr int TC     = CC / CT;        // 256 columns per block
static constexpr int CHROWS = 32;             // rows per LDS chunk
static constexpr int NCH    = PROWS / CHROWS; // 8 chunks, double-buffered
static constexpr int NCOL   = BB * CC;        // 16384 (b,c) columns
static constexpr int WAVES  = 256 / 32;       // wave32
static constexpr int WR     = CHROWS / WAVES; // 4 rows per wave's TDM slice

typedef unsigned int u32x4 __attribute__((ext_vector_type(4)));
typedef int          i32x4 __attribute__((ext_vector_type(4)));
typedef int          i32x8 __attribute__((ext_vector_type(8)));
typedef float        f32x4 __attribute__((ext_vector_type(4)));  // native vec for NT stores

#if __has_builtin(__builtin_amdgcn_tensor_load_to_lds)
#define HAVE_TDM 1
#else
#define HAVE_TDM 0
#endif

// Sorted (descending) top-K insertion: compare/swap chain, fully unrolled.
__device__ __forceinline__ void topk_insert(float (&vals)[KK], int (&idxs)[KK],
                                            float v, int p) {
    if (v <= vals[KK - 1]) return;   // below current 16th-best: reject fast
    float nv = v; int ni = p;
#pragma unroll
    for (int j = 0; j < KK; ++j) {
        bool  gt = nv > vals[j];
        float tv = gt ? vals[j] : nv;
        int   ti = gt ? idxs[j] : ni;
        vals[j]  = gt ? nv : vals[j];
        idxs[j]  = gt ? ni : idxs[j];
        nv = tv; ni = ti;
    }
}

#if HAVE_TDM
// Issue one TDM 2D tile load: TC f32 per row, WR rows, row stride CC elems.
// D# packing per CDNA5 ISA.
__device__ __forceinline__ void tdm_issue(uint64_t gaddr, uint32_t laddr) {
    u32x4 g0;
    g0[0] = 1u;                                            // count=1, user D#
    g0[1] = laddr;                                         // lds_addr [63:32]
    g0[2] = (uint32_t)gaddr;                               // global_addr lo
    g0[3] = ((uint32_t)(gaddr >> 32) & 0x01FFFFFFu)        // global_addr hi (57b)
          | (2u << 30);                                    // type=2 ("image")
    i32x8 g1;
    g1[0] = 0x00020000;                    // data_size=2 -> 4 bytes; no iterate/pad
    g1[1] = (int)((CC & 0xFFFF) << 16);    // tensor_dim0=512  -> bits [79:48]
    g1[2] = (int)(((unsigned)CC >> 16) | ((PP & 0xFFFF) << 16)); // tensor_dim1=4096
    g1[3] = (int)(((unsigned)PP >> 16) | ((unsigned)TC << 16));  // tile_dim0=256
    g1[4] = WR;                            // tile_dim1=4, tile_dim2=0
    g1[5] = CC;                            // tensor_dim0_stride=512 (low 32)
    g1[6] = 0;                             // stride hi16 | dim1_stride lo16
    g1[7] = 0;
    i32x4 gz = {0, 0, 0, 0};
#if defined(__clang_major__) && (__clang_major__ >= 23)
    i32x8 gz8 = {0, 0, 0, 0, 0, 0, 0, 0};
    __builtin_amdgcn_tensor_load_to_lds(g0, g1, gz, gz, gz8, 0);
#else
    __builtin_amdgcn_tensor_load_to_lds(g0, g1, gz, gz, 0);
#endif
}
#endif

// ---------------------------------------------------------------------------
// K1: per-block partial top-16. Block = (b, c-tile, p-split); 1 thread/column.
// TDM streams 32-row x 256-col f32 tiles into double-buffered LDS.
// ---------------------------------------------------------------------------
__global__ __launch_bounds__(256) void topk_partial_kernel(
        const float* __restrict__ sim,
        float* __restrict__ scratchV, int* __restrict__ scratchI) {
    __shared__ float tile[2][CHROWS * TC];   // 2 x 32 KiB

    const int tid = threadIdx.x;
    const int b  = blockIdx.x;
    const int ct = blockIdx.y;
    const int ps = blockIdx.z;
    const int c0 = ct * TC;
    const int p0 = ps * PROWS;

    float vals[KK]; int idxs[KK];
#pragma unroll
    for (int j = 0; j < KK; ++j) { vals[j] = -__builtin_inff(); idxs[j] = 0; }

#if HAVE_TDM
    const int wave = __builtin_amdgcn_readfirstlane(tid >> 5);
    const uint64_t baseByte = (uint64_t)(uintptr_t)sim
        + 4ull * ((uint64_t)b * PP * CC + (uint64_t)p0 * CC + (uint64_t)c0);
    const uint32_t ldsBase = (uint32_t)(uintptr_t)&tile[0][0];

    // chunk 0 in flight before the loop
    tdm_issue(baseByte + 4ull * CC * (uint64_t)(wave * WR),
              ldsBase + (uint32_t)(wave * WR * TC * 4));

    for (int ch = 0; ch < NCH; ++ch) {
        if (ch + 1 < NCH) {
            tdm_issue(baseByte + 4ull * CC * (uint64_t)((ch + 1) * CHROWS + wave * WR),
                      ldsBase + (uint32_t)(((ch + 1) & 1) * (CHROWS * TC * 4)
                                           + wave * WR * TC * 4));
            __builtin_amdgcn_s_wait_tensorcnt((short)1);  // current chunk landed
        } else {
            __builtin_amdgcn_s_wait_tensorcnt((short)0);
        }
        __syncthreads();                                  // all waves' slices ready
        const float* buf = &tile[ch & 1][0];
        const int pbase = p0 + ch * CHROWS;
        for (int r = 0; r < CHROWS; ++r) {
            float v = buf[r * TC + tid];                  // bank-conflict free
            topk_insert(vals, idxs, v, pbase + r);
        }
        __syncthreads();   // done reading: next iter may overwrite other buffer
    }
#else
    // Fallback: cooperative global->LDS staging with plain loads.
    for (int ch = 0; ch < NCH; ++ch) {
        __syncthreads();
        const int pbase = p0 + ch * CHROWS;
        for (int i = tid; i < CHROWS * TC; i += 256) {
            int r = i >> 8;           // TC == 256
            int c = i & (TC - 1);
            tile[0][i] = sim[(size_t)b * PP * CC + (size_t)(pbase + r) * CC + c0 + c];
        }
        __syncthreads();
        for (int r = 0; r < CHROWS; ++r)
            topk_insert(vals, idxs, tile[0][r * TC + tid], pbase + r);
    }
#endif

    // Column-major partials: coalesced, non-temporal (one-shot producer).
    const int col = b * CC + c0 + tid;
#pragma unroll
    for (int j = 0; j < KK; ++j) {
        __builtin_nontemporal_store(vals[j], &scratchV[(ps * KK + j) * NCOL + col]);
        __builtin_nontemporal_store(idxs[j], &scratchI[(ps * KK + j) * NCOL + col]);
    }
}

// ---------------------------------------------------------------------------
// K2: merge 16 partial lists per column -> score + final 16 indices.
// ---------------------------------------------------------------------------
__global__ __launch_bounds__(256) void topk_merge_kernel(
        const float* __restrict__ scratchV, const int* __restrict__ scratchI,
        float* __restrict__ scores, int* __restrict__ topIdx) {
    const int col = blockIdx.x * 256 + threadIdx.x;
    float vals[KK]; int idxs[KK];
#pragma unroll
    for (int j = 0; j < KK; ++j) { vals[j] = -__builtin_inff(); idxs[j] = 0; }
    for (int s = 0; s < PS * KK; ++s) {
        float v = __builtin_nontemporal_load(&scratchV[s * NCOL + col]);
        int   i = __builtin_nontemporal_load(&scratchI[s * NCOL + col]);
        topk_insert(vals, idxs, v, i);
    }
    scores[col] = vals[0];                  // max == max of top-16
#pragma unroll
    for (int j = 0; j < KK; ++j) topIdx[j * NCOL + col] = idxs[j];
}

// ---------------------------------------------------------------------------
// K3: zero the 256 MiB mask with non-temporal 128-bit stores (grid-stride).
// ---------------------------------------------------------------------------
__global__ __launch_bounds__(256) void zero_mask_kernel(f32x4* __restrict__ m4,
                                                        long n4) {
    long i      = (long)blockIdx.x * blockDim.x + threadIdx.x;
    long stride = (long)gridDim.x * blockDim.x;
    f32x4 z = {0.f, 0.f, 0.f, 0.f};
    for (; i < n4; i += stride) __builtin_nontemporal_store(z, &m4[i]);
}

// ---------------------------------------------------------------------------
// K4: scatter 16 ones per column (262k 4B stores, negligible). Regular TH so
// scattered writes to shared cachelines merge in L2.
// ---------------------------------------------------------------------------
__global__ __launch_bounds__(256) void scatter_kernel(
        const int* __restrict__ topIdx, float* __restrict__ mask) {
    const int col = blockIdx.x * 256 + threadIdx.x;
    const int b = col / CC, c = col % CC;
    const size_t base = (size_t)b * PP * CC + (size_t)c;
#pragma unroll
    for (int j = 0; j < KK; ++j) {
        int p = topIdx[j * NCOL + col];
        mask[base + (size_t)p * CC] = 1.0f;
    }
}

extern "C" void kernel_launch(void* const* d_in, const int* in_sizes, int n_in,
                              void* d_out, int out_size, void* d_ws, size_t ws_size,
                              hipStream_t stream) {
    (void)in_sizes; (void)n_in; (void)out_size; (void)ws_size;
    const float* sim = (const float*)d_in[0];
    float* out    = (float*)d_out;
    float* scores = out;               // [B, C] = 16384 f32
    float* mask   = out + NCOL;        // [B, P, C] = 67,108,864 f32

    // Partial lists live temporarily inside the (later zeroed) mask region:
    // 2 x PS*KK*NCOL dwords = 33.5 MB << 256 MB.
    float* scratchV = mask;
    int*   scratchI = (int*)(mask + (size_t)PS * KK * NCOL);
    // Final indices must survive the zero pass -> d_ws (KK*NCOL*4 = 1 MiB).
    int*   topIdx   = (int*)d_ws;

    topk_partial_kernel<<<dim3(BB, CT, PS), 256, 0, stream>>>(sim, scratchV, scratchI);
    topk_merge_kernel<<<NCOL / 256, 256, 0, stream>>>(scratchV, scratchI, scores, topIdx);
    long n4 = (long)BB * PP * CC / 4;
    zero_mask_kernel<<<4096, 256, 0, stream>>>((f32x4*)mask, n4);
    scatter_kernel<<<NCOL / 256, 256, 0, stream>>>(topIdx, mask);
}